// ParameterDecoder_51462298141144
// MI455X (gfx1250) — compile-verified
//
#include <hip/hip_runtime.h>
#include <hip/hip_bf16.h>

// ----------------------------------------------------------------------------
// ParameterDecoder on MI455X (gfx1250, wave32).
// GEMMs on V_WMMA_F32_16X16X4_F32 (fp32 in, fp32 accumulate, matches ref).
// Nc=192 GEMMs (the dominant ones) use a block-cooperative kernel with the
// weight slab staged in LDS via GLOBAL_LOAD_ASYNC_TO_LDS_B128 (ASYNCcnt,
// inline asm per CDNA5 ISA). _sal is algebraically reduced
// to aggregate-then-project: one [1024,576]@[576,192] WMMA GEMM per _sal.
// ----------------------------------------------------------------------------

#define Bsz 32
#define Nn  32
#define Hd  192
#define GEd 512
#define NHd 6144
#define BN  1024      // Bsz*Nn
#define BNN 32768     // Bsz*Nn*Nn

typedef float v2f __attribute__((ext_vector_type(2)));
typedef float v8f __attribute__((ext_vector_type(8)));

// ---- CDNA5 async global->LDS staging (inline asm; set to 0 for plain LDS) --
#define USE_ASYNC_ASM 1

__device__ __forceinline__ void cp16_to_lds(const float* g, float* l) {
#if USE_ASYNC_ASM
  const unsigned int lofs = (unsigned int)(uintptr_t)l;       // LDS byte addr
  const unsigned long long ga = (unsigned long long)(uintptr_t)g;
  asm volatile("global_load_async_to_lds_b128 %0, %1, off"
               :: "v"(lofs), "v"(ga) : "memory");
#else
  *(float4*)l = *(const float4*)g;
#endif
}

__device__ __forceinline__ void wait_async0() {
#if USE_ASYNC_ASM
#if __has_builtin(__builtin_amdgcn_s_wait_asynccnt)
  __builtin_amdgcn_s_wait_asynccnt(0);
#else
  asm volatile("s_wait_asynccnt 0x0" ::: "memory");
#endif
#endif
}

__device__ __forceinline__ float waveSum(float v) {
#pragma unroll
  for (int o = 16; o > 0; o >>= 1) v += __shfl_xor(v, o, 32);
  return v;
}
__device__ __forceinline__ float waveMax(float v) {
#pragma unroll
  for (int o = 16; o > 0; o >>= 1) v = fmaxf(v, __shfl_xor(v, o, 32));
  return v;
}

// ---------------------------------------------------------------------------
// Generic fp32 WMMA GEMM (used for the two context-projection GEMMs).
// One wave per 16x16 output tile; K advanced 4 per v_wmma_f32_16x16x4_f32.
//   A 16x4 : lanes 0-15 -> M=lane, K={k0,k0+1}; lanes 16-31 -> K={k0+2,k0+3}
//   B 4x16 : lanes 0-15 -> N=lane, same K split
//   C 16x16: vgpr p -> row p (lanes 0-15) / p+8 (lanes 16-31), col lane&15
// ---------------------------------------------------------------------------
__global__ __launch_bounds__(128) void k_gemm(const float* __restrict__ A,
                                              const float* __restrict__ W,
                                              const float* __restrict__ bias,
                                              float* __restrict__ C,
                                              int M, int K, int Nc, int relu) {
  const int wave = blockIdx.x * 4 + (threadIdx.x >> 5);
  const int Nt = Nc >> 4;
  const int Mt = M >> 4;
  if (wave >= Mt * Nt) return;
  const int mt = wave / Nt, nt = wave % Nt;
  const int m0 = mt << 4, n0 = nt << 4;
  const int lane = threadIdx.x & 31;
  const int r  = lane & 15;
  const int kh = lane >> 4;

  const float* __restrict__ Arow = A + (long)(m0 + r) * K;
  const float* __restrict__ Wcol = W + n0 + r;

  v8f acc = {};
  for (int k0 = 0; k0 < K; k0 += 4) {
    const int ka = k0 + (kh << 1);
    v2f a, b;
    a.x = Arow[ka];
    a.y = Arow[ka + 1];
    b.x = Wcol[(long)ka * Nc];
    b.y = Wcol[(long)(ka + 1) * Nc];
    acc = __builtin_amdgcn_wmma_f32_16x16x4_f32(false, a, false, b,
                                                (short)0, acc, false, false);
  }
  const float bb = bias ? bias[n0 + r] : 0.0f;
#pragma unroll
  for (int p = 0; p < 8; ++p) {
    float v = acc[p] + bb;
    if (relu) v = fmaxf(v, 0.0f);
    C[(long)(m0 + p + kh * 8) * Nc + n0 + r] = v;
  }
}

// ---------------------------------------------------------------------------
// Block-cooperative fp32 WMMA GEMM specialized for Nc == 192 (K % 16 == 0).
// Block = 128 threads (4 waves); each wave computes 16 rows x 192 cols
// (12 v8f accumulators). The 16x192 weight slab per K-block (12 KB,
// contiguous) is staged in LDS (stride 208 to split bank ranges between the
// two lane halves) and shared by all 4 waves — async-copied to LDS.
// ---------------------------------------------------------------------------
#define SB_STRIDE 208
__global__ __launch_bounds__(128) void k_gemm_n192(const float* __restrict__ A,
                                                   const float* __restrict__ W,
                                                   const float* __restrict__ bias,
                                                   float* __restrict__ C,
                                                   int M, int K, int relu) {
  __shared__ __align__(16) float sB[16 * SB_STRIDE];  // 13 KB
  const int tid  = threadIdx.x;
  const int wv   = tid >> 5;
  const int lane = tid & 31;
  const int r    = lane & 15;
  const int kh   = lane >> 4;
  const int m0   = blockIdx.x * 64 + wv * 16;

  v8f acc[12] = {};
  const float* __restrict__ Arow = A + (long)(m0 + r) * K;

  for (int k0 = 0; k0 < K; k0 += 16) {
    __syncthreads();  // previous slab fully consumed
    {
      const float* src = W + (long)k0 * Hd;
#pragma unroll
      for (int i = 0; i < 6; ++i) {          // 768 float4 chunks / 128 threads
        const int idx4 = tid + i * 128;
        const int row  = idx4 / 48;          // K row within slab
        const int c4   = idx4 % 48;          // float4 column
        cp16_to_lds(src + row * Hd + c4 * 4, &sB[row * SB_STRIDE + c4 * 4]);
      }
    }
    wait_async0();
    __syncthreads();  // slab visible to all waves

#pragma unroll
    for (int kk = 0; kk < 16; kk += 4) {
      const int ka = kk + (kh << 1);
      v2f a;
      a.x = Arow[k0 + ka];
      a.y = Arow[k0 + ka + 1];
#pragma unroll
      for (int nt = 0; nt < 12; ++nt) {
        v2f b;
        b.x = sB[ka * SB_STRIDE + nt * 16 + r];
        b.y = sB[(ka + 1) * SB_STRIDE + nt * 16 + r];
        acc[nt] = __builtin_amdgcn_wmma_f32_16x16x4_f32(false, a, false, b,
                                                        (short)0, acc[nt],
                                                        false, false);
      }
    }
  }

#pragma unroll
  for (int nt = 0; nt < 12; ++nt) {
    const float bb = bias ? bias[nt * 16 + r] : 0.0f;
#pragma unroll
    for (int p = 0; p < 8; ++p) {
      float v = acc[nt][p] + bb;
      if (relu) v = fmaxf(v, 0.0f);
      C[(long)(m0 + p + kh * 8) * Hd + nt * 16 + r] = v;
    }
  }
}

// nt normalization (sigmoid ch0, softmax ch1..5) + node_info = relu(nt@en_w+b)
__global__ __launch_bounds__(192) void k_node_info(const float* __restrict__ nt,
                                                   const float* __restrict__ enw,
                                                   const float* __restrict__ enb,
                                                   float* __restrict__ Nf) {
  const int rrow = blockIdx.x;  // 0..1023
  const int n = threadIdx.x;
  __shared__ float s[6], c[6];
  if (n < 6) s[n] = nt[rrow * 6 + n];
  __syncthreads();
  if (n == 0) {
    float mx = s[1];
    for (int i = 2; i < 6; ++i) mx = fmaxf(mx, s[i]);
    float e[5], sum = 0.0f;
    for (int i = 0; i < 5; ++i) { e[i] = __expf(s[1 + i] - mx); sum += e[i]; }
    c[0] = 1.0f / (1.0f + __expf(-s[0]));
    const float inv = 1.0f / sum;
    for (int i = 0; i < 5; ++i) c[1 + i] = e[i] * inv;
  }
  __syncthreads();
  float acc = enb[n];
#pragma unroll
  for (int i = 0; i < 6; ++i) acc += c[i] * enw[i * Hd + n];
  Nf[(long)rrow * Hd + n] = fmaxf(acc, 0.0f);
}

// et normalization (3 softmaxes of 4) + edge_info = relu(et@ee_w+b)
__global__ __launch_bounds__(192) void k_edge_info(const float* __restrict__ et,
                                                   const float* __restrict__ eew,
                                                   const float* __restrict__ eeb,
                                                   float* __restrict__ E) {
  const long rrow = blockIdx.x;  // 0..32767
  const int n = threadIdx.x;
  __shared__ float s[12], c[12];
  if (n < 12) s[n] = et[rrow * 12 + n];
  __syncthreads();
  if (n < 3) {
    const int g = n * 4;
    float mx = fmaxf(fmaxf(s[g], s[g + 1]), fmaxf(s[g + 2], s[g + 3]));
    float e0 = __expf(s[g] - mx), e1 = __expf(s[g + 1] - mx);
    float e2 = __expf(s[g + 2] - mx), e3 = __expf(s[g + 3] - mx);
    const float inv = 1.0f / (e0 + e1 + e2 + e3);
    c[g] = e0 * inv; c[g + 1] = e1 * inv; c[g + 2] = e2 * inv; c[g + 3] = e3 * inv;
  }
  __syncthreads();
  float acc = eeb[n];
#pragma unroll
  for (int i = 0; i < 12; ++i) acc += c[i] * eew[i * Hd + n];
  E[rrow * Hd + n] = fmaxf(acc, 0.0f);
}

// la_i[r] = Nf[r]·wa[0:192], la_j[r] = Nf[r]·wa[192:384]
__global__ __launch_bounds__(192) void k_node_logit(const float* __restrict__ Nf,
                                                    const float* __restrict__ wa,
                                                    float* __restrict__ la_i,
                                                    float* __restrict__ la_j) {
  const int rrow = blockIdx.x;
  const int t = threadIdx.x;
  const float x = Nf[(long)rrow * Hd + t];
  __shared__ float rA[6], rB[6];
  const float sa = waveSum(x * wa[t]);
  const float sb = waveSum(x * wa[Hd + t]);
  if ((t & 31) == 0) { rA[t >> 5] = sa; rB[t >> 5] = sb; }
  __syncthreads();
  if (t == 0) {
    float A2 = 0.0f, B2 = 0.0f;
    for (int i = 0; i < 6; ++i) { A2 += rA[i]; B2 += rB[i]; }
    la_i[rrow] = A2; la_j[rrow] = B2;
  }
}

// el[r] = E[r]·wa[384:576], one wave per edge row
__global__ __launch_bounds__(256) void k_edge_logit(const float* __restrict__ E,
                                                    const float* __restrict__ wae,
                                                    float* __restrict__ el, int rows) {
  const int rrow = blockIdx.x * 8 + (threadIdx.x >> 5);
  if (rrow >= rows) return;
  const int lane = threadIdx.x & 31;
  const float* row = E + (long)rrow * Hd;
  float s = 0.0f;
#pragma unroll
  for (int c = 0; c < 6; ++c) s += row[lane + c * 32] * wae[lane + c * 32];
  s = waveSum(s);
  if (lane == 0) el[rrow] = s;
}

// softmax over j of logits[b,i,j] = la_i + la_j + el + ba; one wave per (b,i)
__global__ __launch_bounds__(256) void k_softmax(const float* __restrict__ la_i,
                                                 const float* __restrict__ la_j,
                                                 const float* __restrict__ el,
                                                 const float* __restrict__ ba,
                                                 float* __restrict__ wsm) {
  const int ri = blockIdx.x * 8 + (threadIdx.x >> 5);  // 0..1023
  const int j = threadIdx.x & 31;
  const int b = ri >> 5;
  float lg = la_i[ri] + la_j[(b << 5) + j] + el[((long)ri << 5) + j] + ba[0];
  const float mx = waveMax(lg);
  const float e = __expf(lg - mx);
  const float sm = waveSum(e);
  wsm[((long)ri << 5) + j] = e / sm;
}

// salA[ri] = [ Nf[b,i] | Σ_j w·Nf[b,j] | Σ_j w·E[b,i,j] ]  (1024 x 576)
__global__ __launch_bounds__(192) void k_build_salA(const float* __restrict__ Nf,
                                                    const float* __restrict__ E,
                                                    const float* __restrict__ wsm,
                                                    float* __restrict__ salA) {
  const int ri = blockIdx.x;
  const int n = threadIdx.x;
  const int b = ri >> 5;
  __shared__ float wl[32];
  if (n < 32) wl[n] = wsm[((long)ri << 5) + n];
  __syncthreads();
  const float* Eb = E + ((long)ri << 5) * Hd + n;
  const float* Nb = Nf + (long)(b << 5) * Hd + n;
  float sN = 0.0f, sE = 0.0f;
#pragma unroll
  for (int j = 0; j < 32; ++j) {
    const float w = wl[j];
    sN += w * Nb[(long)j * Hd];
    sE += w * Eb[(long)j * Hd];
  }
  float* dst = salA + (long)ri * 576;
  dst[n]       = Nf[(long)ri * Hd + n];
  dst[192 + n] = sN;
  dst[384 + n] = sE;
}

// P = LN(newp*(1+salmul) + saladd + P) * g + bt   (row-wise, H=192)
__global__ __launch_bounds__(192) void k_film_ln(const float* __restrict__ newp,
                                                 const float* __restrict__ sm,
                                                 const float* __restrict__ sa,
                                                 float* __restrict__ P,
                                                 const float* __restrict__ g,
                                                 const float* __restrict__ bt) {
  const int rrow = blockIdx.x;
  const int t = threadIdx.x;
  const long idx = (long)rrow * Hd + t;
  const float np = newp[idx];
  const float x = np * sm[idx] + np + sa[idx] + P[idx];
  __shared__ float rs[6], rq[6], mv[2];
  const float s1 = waveSum(x), s2 = waveSum(x * x);
  if ((t & 31) == 0) { rs[t >> 5] = s1; rq[t >> 5] = s2; }
  __syncthreads();
  if (t == 0) {
    float S = 0.0f, Q = 0.0f;
    for (int i = 0; i < 6; ++i) { S += rs[i]; Q += rq[i]; }
    const float m = S / (float)Hd;
    mv[0] = m;
    mv[1] = rsqrtf(Q / (float)Hd - m * m + 1e-5f);
  }
  __syncthreads();
  P[idx] = (x - mv[0]) * mv[1] * g[t] + bt[t];
}

// out = LN(a + out) * g + bt  (in-place on `out`, row-wise)
__global__ __launch_bounds__(192) void k_add_ln(const float* __restrict__ a,
                                                float* __restrict__ out,
                                                const float* __restrict__ g,
                                                const float* __restrict__ bt) {
  const int rrow = blockIdx.x;
  const int t = threadIdx.x;
  const long idx = (long)rrow * Hd + t;
  const float x = a[idx] + out[idx];
  __shared__ float rs[6], rq[6], mv[2];
  const float s1 = waveSum(x), s2 = waveSum(x * x);
  if ((t & 31) == 0) { rs[t >> 5] = s1; rq[t >> 5] = s2; }
  __syncthreads();
  if (t == 0) {
    float S = 0.0f, Q = 0.0f;
    for (int i = 0; i < 6; ++i) { S += rs[i]; Q += rq[i]; }
    const float m = S / (float)Hd;
    mv[0] = m;
    mv[1] = rsqrtf(Q / (float)Hd - m * m + 1e-5f);
  }
  __syncthreads();
  out[idx] = (x - mv[0]) * mv[1] * g[t] + bt[t];
}

// Final head: xr = relu(x*(1+sm)+sa); out[r,0:8] = xr @ out_w + out_b
__global__ __launch_bounds__(192) void k_out(const float* __restrict__ x,
                                             const float* __restrict__ sm,
                                             const float* __restrict__ sa,
                                             const float* __restrict__ ow,
                                             const float* __restrict__ ob,
                                             float* __restrict__ out) {
  const int rrow = blockIdx.x;
  const int t = threadIdx.x;
  const long idx = (long)rrow * Hd + t;
  __shared__ float xr[Hd];
  const float v = x[idx];
  xr[t] = fmaxf(v * sm[idx] + v + sa[idx], 0.0f);
  __syncthreads();
  if (t < 8) {
    float acc = ob[t];
    for (int i = 0; i < Hd; ++i) acc += xr[i] * ow[i * 8 + t];
    out[(long)rrow * 8 + t] = acc;
  }
}

// ---------------------------------------------------------------------------
static inline void launch_gemm(const float* A, const float* W, const float* bias,
                               float* C, int M, int K, int Nc, int relu,
                               hipStream_t s) {
  const int tiles = (M >> 4) * (Nc >> 4);
  const int blocks = (tiles + 3) >> 2;
  hipLaunchKernelGGL(k_gemm, dim3(blocks), dim3(128), 0, s, A, W, bias, C, M, K, Nc, relu);
}

static inline void launch_gemm192(const float* A, const float* W, const float* bias,
                                  float* C, int M, int K, int relu, hipStream_t s) {
  hipLaunchKernelGGL(k_gemm_n192, dim3(M / 64), dim3(128), 0, s, A, W, bias, C, M, K, relu);
}

static inline void run_sal(const float* Nf, const float* E,
                           const float* wa, const float* ba,
                           const float* wv, const float* bv, float* out,
                           float* la_i, float* la_j, float* el, float* wsm,
                           float* salA, hipStream_t s) {
  hipLaunchKernelGGL(k_node_logit, dim3(BN), dim3(192), 0, s, Nf, wa, la_i, la_j);
  hipLaunchKernelGGL(k_edge_logit, dim3(BNN / 8), dim3(256), 0, s, E, wa + 384, el, BNN);
  hipLaunchKernelGGL(k_softmax, dim3(BN / 8), dim3(256), 0, s, la_i, la_j, el, ba, wsm);
  hipLaunchKernelGGL(k_build_salA, dim3(BN), dim3(192), 0, s, Nf, E, wsm, salA);
  launch_gemm192(salA, wv, bv, out, BN, 576, 0, s);
}

// Flattened input order (depth-first insertion order of setup_inputs dict):
// 0 graph_embs, 1 node_types, 2 edge_types,
// 3 cp_w1, 4 cp_b1, 5 cp_w2, 6 cp_b2, 7 en_w, 8 en_b, 9 ee_w, 10 ee_b,
// per layer l (base = 11 + 20*l):
//   +0 lp_w, +1 lp_b, +2..5 film_mul{wa,ba,wv,bv}, +6..9 film_add{wa,ba,wv,bv},
//   +10 ln_w, +11 ln_b, +12 le_w, +13 le_b, +14 p_g, +15 p_bt, +16 n_g,
//   +17 n_bt, +18 e_g, +19 e_bt
// 171 fp_w, 172 fp_b, 173..176 f_mul, 177..180 f_add, 181 out_w, 182 out_b.
extern "C" void kernel_launch(void* const* d_in, const int* in_sizes, int n_in,
                              void* d_out, int out_size, void* d_ws, size_t ws_size,
                              hipStream_t stream) {
  (void)in_sizes; (void)n_in; (void)out_size; (void)ws_size;
#define IN(i) ((const float*)d_in[i])

  float* ws = (float*)d_ws;
  float* h1     = ws;              ws += Bsz * NHd;   // 196608
  float* P      = ws;              ws += BN * Hd;
  float* Nf     = ws;              ws += BN * Hd;
  float* newp   = ws;              ws += BN * Hd;
  float* newn   = ws;              ws += BN * Hd;
  float* salmul = ws;              ws += BN * Hd;
  float* saladd = ws;              ws += BN * Hd;
  float* salA   = ws;              ws += BN * 576;
  float* la_i   = ws;              ws += BN;
  float* la_j   = ws;              ws += BN;
  float* el     = ws;              ws += BNN;
  float* wsm    = ws;              ws += BNN;
  float* E      = ws;              ws += BNN * Hd;    // 25 MB
  float* newe   = ws;              ws += BNN * Hd;    // 25 MB

  // --- context projection: node_params seeds ---
  launch_gemm(IN(0), IN(3), IN(4), h1, Bsz, GEd, NHd, 1, stream);
  launch_gemm(h1, IN(5), IN(6), P, Bsz, NHd, NHd, 1, stream);  // P == [1024,192] flat

  // --- input encoders ---
  hipLaunchKernelGGL(k_node_info, dim3(BN), dim3(192), 0, stream, IN(1), IN(7), IN(8), Nf);
  hipLaunchKernelGGL(k_edge_info, dim3(BNN), dim3(192), 0, stream, IN(2), IN(9), IN(10), E);

  // --- 8 message-passing layers ---
  for (int l = 0; l < 8; ++l) {
    const int base = 11 + 20 * l;
    launch_gemm192(P, IN(base + 0), IN(base + 1), newp, BN, Hd, 0, stream);
    run_sal(Nf, E, IN(base + 2), IN(base + 3), IN(base + 4), IN(base + 5),
            salmul, la_i, la_j, el, wsm, salA, stream);
    run_sal(Nf, E, IN(base + 6), IN(base + 7), IN(base + 8), IN(base + 9),
            saladd, la_i, la_j, el, wsm, salA, stream);
    launch_gemm192(Nf, IN(base + 10), IN(base + 11), newn, BN, Hd, 0, stream);
    launch_gemm192(E, IN(base + 12), IN(base + 13), newe, BNN, Hd, 0, stream);
    hipLaunchKernelGGL(k_film_ln, dim3(BN), dim3(192), 0, stream,
                       newp, salmul, saladd, P, IN(base + 14), IN(base + 15));
    hipLaunchKernelGGL(k_add_ln, dim3(BN), dim3(192), 0, stream,
                       newn, Nf, IN(base + 16), IN(base + 17));
    hipLaunchKernelGGL(k_add_ln, dim3(BNN), dim3(192), 0, stream,
                       newe, E, IN(base + 18), IN(base + 19));
  }

  // --- final head ---
  launch_gemm192(P, IN(171), IN(172), newp, BN, Hd, 0, stream);
  run_sal(Nf, E, IN(173), IN(174), IN(175), IN(176), salmul,
          la_i, la_j, el, wsm, salA, stream);
  run_sal(Nf, E, IN(177), IN(178), IN(179), IN(180), saladd,
          la_i, la_j, el, wsm, salA, stream);
  hipLaunchKernelGGL(k_out, dim3(BN), dim3(192), 0, stream,
                     newp, salmul, saladd, IN(181), IN(182), (float*)d_out);
#undef IN
}